// AE_3_66632122630507
// MI455X (gfx1250) — compile-verified
//
#include <hip/hip_runtime.h>

// f(x) = c*x + w * k * tanh(x@w / D + b)   for x[B, D], D = 4096, B = 16384
// Memory-bound: ~512 MB of HBM traffic -> ~22 us floor at 23.3 TB/s.
// GEMV dot-products computed with V_WMMA_F32_16X16X4_F32 (free under roofline),
// w staged to LDS via gfx1250 async global->LDS loads.

#define D_DIM     4096
#define TILE_ROWS 16
#define THREADS   256
#define NWAVES    (THREADS / 32)          // 8 waves (wave32)
#define K_SLICE   (D_DIM / NWAVES)        // 512 columns per wave

typedef float v2f __attribute__((ext_vector_type(2)));
typedef float v4f __attribute__((ext_vector_type(4)));
typedef float v8f __attribute__((ext_vector_type(8)));
typedef int   v4i __attribute__((ext_vector_type(4)));

#if defined(__gfx1250__) && \
    __has_builtin(__builtin_amdgcn_global_load_async_to_lds_b128) && \
    __has_builtin(__builtin_amdgcn_s_wait_asynccnt)
#define HAVE_ASYNC 1
// Builtin signature (from hipcc diagnostic): param0 = v4i in device (AS1) space,
// param1 = v4i in LDS (AS3) space, then imm offset, imm cpol.
typedef __attribute__((address_space(1))) v4i* g_v4i_p;
typedef __attribute__((address_space(3))) v4i* l_v4i_p;
#else
#define HAVE_ASYNC 0
#endif

__global__ __launch_bounds__(THREADS)
void tanh_rank1_wmma_kernel(const float* __restrict__ x,
                            const float* __restrict__ w,
                            const float* __restrict__ cp,
                            const float* __restrict__ kp,
                            const float* __restrict__ bp,
                            float* __restrict__ y) {
    __shared__ float s_w[D_DIM];                   // 16 KB: w staged once per block
    __shared__ float s_part[NWAVES * TILE_ROWS];   // per-wave partial dots
    __shared__ float s_h[TILE_ROWS];               // h per row

    const int tid  = threadIdx.x;
    const int row0 = blockIdx.x * TILE_ROWS;

    // ---- Phase 0: stage w into LDS with async global->LDS copies ----
    // each thread moves 4 x float4 = 16 floats; 256 threads cover 4096 floats
    #pragma unroll
    for (int i = 0; i < 4; ++i) {
        const int idx = (i * THREADS + tid) * 4;
#if HAVE_ASYNC
        __builtin_amdgcn_global_load_async_to_lds_b128(
            (g_v4i_p)(w + idx), (l_v4i_p)(&s_w[idx]), 0, 0);
#else
        *(v4f*)&s_w[idx] = *(const v4f*)(w + idx);
#endif
    }
#if HAVE_ASYNC
    __builtin_amdgcn_s_wait_asynccnt(0);
#endif
    __syncthreads();

    const float cc = cp[0];
    const float kk = kp[0];
    const float bb = bp[0];

    // ---- Phase 1: dot(row, w) via V_WMMA_F32_16X16X4_F32 ----
    // A (16x4 f32, documented layout): lane m (0-15): VGPR0=A[m][0], VGPR1=A[m][1];
    //                                  lane 16+m:     VGPR0=A[m][2], VGPR1=A[m][3].
    // A[m][k] = x[row0+m][kbase+k].  B[k][n] = w[kbase+k] (broadcast over n),
    // so D[m][n] = partial dot of row m, replicated across n; accumulate in C.
    const int lane = tid & 31;
    const int wave = tid >> 5;
    const int m    = lane & 15;
    const int koff = (lane >> 4) << 1;             // 0 for lanes 0-15, 2 for 16-31
    const float* xr = x + (size_t)(row0 + m) * D_DIM + koff;

    v8f acc = {0.f, 0.f, 0.f, 0.f, 0.f, 0.f, 0.f, 0.f};
    const int kbase = wave * K_SLICE;
    for (int k = kbase; k < kbase + K_SLICE; k += 4) {
        v2f a = *(const v2f*)(xr + k);             // per-lane b64, rows striped
        v2f b;                                     // B column-per-lane, K over (vgpr, lane-half)
        b.x = s_w[k + koff];
        b.y = s_w[k + koff + 1];
        acc = __builtin_amdgcn_wmma_f32_16x16x4_f32(
            /*neg_a=*/false, a, /*neg_b=*/false, b,
            /*c_mod=*/(short)0, acc, /*reuse_a=*/false, /*reuse_b=*/false);
    }
    // D layout: lane L(0-15): acc[r] = D[r][L]; lane 16+L: acc[r] = D[8+r][L].
    // D is replicated over columns -> lanes 0 and 16 hold dots for rows 0-7 / 8-15.
    if (m == 0) {
        const int half = lane >> 4;
        #pragma unroll
        for (int r = 0; r < 8; ++r)
            s_part[wave * TILE_ROWS + half * 8 + r] = acc[r];
    }
    __syncthreads();

    if (tid < TILE_ROWS) {
        float dot = 0.f;
        #pragma unroll
        for (int wv = 0; wv < NWAVES; ++wv) dot += s_part[wv * TILE_ROWS + tid];
        s_h[tid] = kk * tanhf(dot * (1.0f / D_DIM) + bb);
    }
    __syncthreads();

    // ---- Phase 2: y = h*w + c*x, coalesced, non-temporal stores ----
    // x tile (256 KB) is L2-hot from phase 1 -> HBM sees x only once.
    for (int r = 0; r < TILE_ROWS; ++r) {
        const float hv = s_h[r];
        const float* xr2 = x + (size_t)(row0 + r) * D_DIM;
        float*       yr  = y + (size_t)(row0 + r) * D_DIM;
        #pragma unroll
        for (int i = 0; i < D_DIM / (THREADS * 4); ++i) {
            const int j = (i * THREADS + tid) * 4;
            v4f xv = __builtin_nontemporal_load((const v4f*)(xr2 + j)); // last use
            v4f wv = *(const v4f*)(&s_w[j]);
            v4f yv;
            yv.x = fmaf(hv, wv.x, cc * xv.x);
            yv.y = fmaf(hv, wv.y, cc * xv.y);
            yv.z = fmaf(hv, wv.z, cc * xv.z);
            yv.w = fmaf(hv, wv.w, cc * xv.w);
            __builtin_nontemporal_store(yv, (v4f*)(yr + j));            // stream out
        }
    }
}

extern "C" void kernel_launch(void* const* d_in, const int* in_sizes, int n_in,
                              void* d_out, int out_size, void* d_ws, size_t ws_size,
                              hipStream_t stream) {
    const float* x  = (const float*)d_in[0];
    const float* w  = (const float*)d_in[1];
    const float* cp = (const float*)d_in[2];
    const float* kp = (const float*)d_in[3];
    const float* bp = (const float*)d_in[4];
    float* y = (float*)d_out;

    const int Dd = in_sizes[1];                // 4096
    const int Bt = in_sizes[0] / Dd;           // 16384
    const int tiles = Bt / TILE_ROWS;          // 1024 blocks

    tanh_rank1_wmma_kernel<<<tiles, THREADS, 0, stream>>>(x, w, cp, kp, bp, y);
}